// GNNStack_61435212202234
// MI455X (gfx1250) — compile-verified
//
#include <hip/hip_runtime.h>
#include <math.h>

typedef __attribute__((ext_vector_type(2))) float v2f;
typedef __attribute__((ext_vector_type(8))) float v8f;

#define CDIM   128
#define EDGES  16
#define NB     20000          // nodes per batch
#define BATCH  4
#define TOTAL  (NB * BATCH)   // 80000 rows
#define WAVES_PER_WG 8
#define TILE_M 16

// ---------------------------------------------------------------------------
// invdeg[g] = 1/sqrt(1 + #valid_edges)
// ---------------------------------------------------------------------------
__global__ __launch_bounds__(256) void k_invdeg(const int* __restrict__ e,
                                                float* __restrict__ idg) {
  int g = blockIdx.x * 256 + threadIdx.x;
  if (g >= TOTAL) return;
  const int4* ep = (const int4*)(e + (size_t)g * EDGES);
  int cnt = 1;
#pragma unroll
  for (int i = 0; i < 4; ++i) {
    int4 v = ep[i];
    cnt += (v.x >= 0) + (v.y >= 0) + (v.z >= 0) + (v.w >= 0);
  }
  idg[g] = rsqrtf((float)cnt);
}

// ---------------------------------------------------------------------------
// H[n,o] = (sum_c X[n,c] * W[o,c]) * invdeg[n]
// One wave -> 16 nodes x 128 outputs via V_WMMA_F32_16X16X4_F32.
// W staged in LDS once per WG with XOR swizzle: Wlds[c*128 + (o ^ (c&63))].
// ---------------------------------------------------------------------------
__global__ __launch_bounds__(256) void k_transform(const float* __restrict__ X,
                                                   const float* __restrict__ W,
                                                   const float* __restrict__ idg,
                                                   float* __restrict__ H) {
  __shared__ float Wlds[CDIM * CDIM];   // 64 KB
  const int t = threadIdx.x;

  // Cooperative coalesced stage of W (row-major [o][c]) into swizzled LDS.
#pragma unroll
  for (int i = 0; i < 16; ++i) {
    int e0 = (i * 256 + t) * 4;                  // 4 consecutive c within one o-row
    const float4 w4 = *(const float4*)(W + e0);
    int o = e0 >> 7;
    int c = e0 & 127;
    Wlds[(c + 0) * CDIM + (o ^ ((c + 0) & 63))] = w4.x;
    Wlds[(c + 1) * CDIM + (o ^ ((c + 1) & 63))] = w4.y;
    Wlds[(c + 2) * CDIM + (o ^ ((c + 2) & 63))] = w4.z;
    Wlds[(c + 3) * CDIM + (o ^ ((c + 3) & 63))] = w4.w;
  }
  __syncthreads();

  const int wave = t >> 5;
  const int lane = t & 31;
  const int half = lane >> 4;     // 0: lanes 0-15, 1: lanes 16-31
  const int l16  = lane & 15;
  const int node_base = (blockIdx.x * WAVES_PER_WG + wave) * TILE_M;

  const float* Xrow = X + (size_t)(node_base + l16) * CDIM;

  v8f acc[8];
#pragma unroll
  for (int nt = 0; nt < 8; ++nt) acc[nt] = (v8f)0.0f;

  // K loop: 128 channels, 4 per WMMA step.
  for (int ks = 0; ks < 32; ++ks) {
    const int c0 = ks * 4 + half * 2;           // ISA 16x4 f32 A layout
    v2f a;
    a.x = Xrow[c0];
    a.y = Xrow[c0 + 1];
#pragma unroll
    for (int nt = 0; nt < 8; ++nt) {
      const int o = nt * 16 + l16;              // B: N = lane, same half split for K
      v2f b;
      b.x = Wlds[(c0    ) * CDIM + (o ^ ((c0    ) & 63))];
      b.y = Wlds[(c0 + 1) * CDIM + (o ^ ((c0 + 1) & 63))];
      acc[nt] = __builtin_amdgcn_wmma_f32_16x16x4_f32(
          /*neg_a=*/false, a, /*neg_b=*/false, b,
          /*c_mod=*/(short)0, acc[nt], /*reuse_a=*/false, /*reuse_b=*/false);
    }
  }

  // C/D layout: VGPR r -> row M = r + 8*half, column N = l16 (+ nt*16).
  float s[8];
#pragma unroll
  for (int r = 0; r < 8; ++r) s[r] = idg[node_base + half * 8 + r];

#pragma unroll
  for (int nt = 0; nt < 8; ++nt) {
#pragma unroll
    for (int r = 0; r < 8; ++r) {
      H[(size_t)(node_base + half * 8 + r) * CDIM + nt * 16 + l16] =
          acc[nt][r] * s[r];
    }
  }
}

// ---------------------------------------------------------------------------
// Y[n] = elu( (H[n] + sum_{e: idx>=0} H[base+idx]) * invdeg[n] )
// One wave per node; lane owns 4 channels (float4). Edge ids broadcast via
// __shfl so the validity branch is wave-uniform.
// ---------------------------------------------------------------------------
__device__ __forceinline__ float elu1(float v) {
  return v > 0.0f ? v : (expf(v) - 1.0f);
}

__global__ __launch_bounds__(256) void k_gather(const float* __restrict__ H,
                                                const int* __restrict__ eidx,
                                                const float* __restrict__ idg,
                                                float* __restrict__ Y) {
  const int gw   = (int)((blockIdx.x * 256 + threadIdx.x) >> 5);  // node id
  const int lane = threadIdx.x & 31;
  const int base = (gw / NB) * NB;                                // batch row base

  float4 acc = *(const float4*)(H + (size_t)gw * CDIM + lane * 4);

  const int ev = eidx[(size_t)gw * EDGES + (lane & 15)];
#pragma unroll
  for (int e = 0; e < EDGES; ++e) {
    const int idx = __shfl(ev, e, 32);          // uniform across the wave
    if (idx >= 0) {
      const float4 nv =
          *(const float4*)(H + (size_t)(base + idx) * CDIM + lane * 4);
      acc.x += nv.x; acc.y += nv.y; acc.z += nv.z; acc.w += nv.w;
    }
  }

  const float s = idg[gw];
  float4 r;
  r.x = elu1(acc.x * s);
  r.y = elu1(acc.y * s);
  r.z = elu1(acc.z * s);
  r.w = elu1(acc.w * s);
  *(float4*)(Y + (size_t)gw * CDIM + lane * 4) = r;
}

// ---------------------------------------------------------------------------
extern "C" void kernel_launch(void* const* d_in, const int* in_sizes, int n_in,
                              void* d_out, int out_size, void* d_ws, size_t ws_size,
                              hipStream_t stream) {
  const float* x     = (const float*)d_in[0];
  const int*   edges = (const int*)d_in[1];
  const float* W0    = (const float*)d_in[2];
  const float* W1    = (const float*)d_in[3];
  const float* W2    = (const float*)d_in[4];
  float* out = (float*)d_out;

  // ws layout: [ invdeg: TOTAL floats | h: TOTAL*128 floats ]
  float* idg = (float*)d_ws;
  size_t idg_bytes = ((size_t)TOTAL * sizeof(float) + 255) & ~(size_t)255;
  float* h = (float*)((char*)d_ws + idg_bytes);

  const dim3 blk(256);
  const dim3 g_deg((TOTAL + 255) / 256);
  const dim3 g_tr(TOTAL / (WAVES_PER_WG * TILE_M));  // 625
  const dim3 g_ga(TOTAL / 8);                        // 8 waves per block -> 10000

  k_invdeg<<<g_deg, blk, 0, stream>>>(edges, idg);

  // Layer 0: x -> h -> out
  k_transform<<<g_tr, blk, 0, stream>>>(x, W0, idg, h);
  k_gather  <<<g_ga, blk, 0, stream>>>(h, edges, idg, out);
  // Layer 1: out -> h -> out
  k_transform<<<g_tr, blk, 0, stream>>>(out, W1, idg, h);
  k_gather  <<<g_ga, blk, 0, stream>>>(h, edges, idg, out);
  // Layer 2: out -> h -> out
  k_transform<<<g_tr, blk, 0, stream>>>(out, W2, idg, h);
  k_gather  <<<g_ga, blk, 0, stream>>>(h, edges, idg, out);
}